// GraphConvolution_3178275799083
// MI455X (gfx1250) — compile-verified
//
#include <hip/hip_runtime.h>

typedef __attribute__((ext_vector_type(2))) float v2f;
typedef __attribute__((ext_vector_type(8))) float v8f;

#define N_NODES 100000
#define N_EDGES 3200000
#define F_IN    256
#define UNITS   256
#define SCAN_B  1024
#define N_SCAN_BLOCKS ((N_NODES + SCAN_B - 1) / SCAN_B)   // 98

// d_ws layout (offsets in 4-byte units)
#define WS_CONV    0                      // float[25,600,000]
#define WS_DEG     25600000               // int[100,000]
#define WS_ROWOFF  25700000               // int[100,001]
#define WS_CURSOR  25800004               // int[100,000]
#define WS_BSUM    25900004               // int[128]
#define WS_SSRC    25900132               // int[3,200,000]
#define WS_SVAL    29100132               // float[3,200,000]
#define WS_WPACK   32300136               // float[65,536] (float2-aligned)

// ---------------------------------------------------------------------------
// 1) zero degree histogram
// ---------------------------------------------------------------------------
__global__ void zero_deg(int* __restrict__ deg) {
    int i = blockIdx.x * blockDim.x + threadIdx.x;
    if (i < N_NODES) deg[i] = 0;
}

// ---------------------------------------------------------------------------
// 2) degree histogram (3.2M int atomics)
// ---------------------------------------------------------------------------
__global__ void count_edges(const int* __restrict__ edst, int* __restrict__ deg) {
    int e = blockIdx.x * blockDim.x + threadIdx.x;
    if (e < N_EDGES) atomicAdd(&deg[edst[e]], 1);
}

// ---------------------------------------------------------------------------
// 3a) per-block exclusive scan (wave32 shuffle scan, 2 barriers/block)
//     writes local-exclusive values to row_off, block totals to bsum
// ---------------------------------------------------------------------------
__global__ void __launch_bounds__(SCAN_B) scan_block(const int* __restrict__ deg,
                                                     int* __restrict__ row_off,
                                                     int* __restrict__ bsum) {
    int i    = blockIdx.x * SCAN_B + threadIdx.x;
    int lane = threadIdx.x & 31;
    int wid  = threadIdx.x >> 5;
    int v    = (i < N_NODES) ? deg[i] : 0;

    // wave-level inclusive scan
    int incl = v;
#pragma unroll
    for (int off = 1; off < 32; off <<= 1) {
        int t = __shfl_up(incl, off, 32);
        if (lane >= off) incl += t;
    }

    __shared__ int wsum[32];
    if (lane == 31) wsum[wid] = incl;
    __syncthreads();

    if (wid == 0) {
        int w  = wsum[lane];
        int wi = w;
#pragma unroll
        for (int off = 1; off < 32; off <<= 1) {
            int t = __shfl_up(wi, off, 32);
            if (lane >= off) wi += t;
        }
        wsum[lane] = wi - w;                       // exclusive wave prefix
        if (lane == 31) bsum[blockIdx.x] = wi;     // block total
    }
    __syncthreads();

    if (i < N_NODES) row_off[i] = (incl - v) + wsum[wid];
}

// ---------------------------------------------------------------------------
// 3b) scan the 98 block totals (serial: trivial size)
// ---------------------------------------------------------------------------
__global__ void scan_partials(int* __restrict__ bsum, int* __restrict__ row_off) {
    if (threadIdx.x == 0 && blockIdx.x == 0) {
        int running = 0;
        for (int i = 0; i < N_SCAN_BLOCKS; ++i) {
            int t = bsum[i];
            bsum[i] = running;
            running += t;
        }
        row_off[N_NODES] = running;                // == N_EDGES
    }
}

// ---------------------------------------------------------------------------
// 3c) apply block offsets; initialize cursor
// ---------------------------------------------------------------------------
__global__ void __launch_bounds__(SCAN_B) scan_apply(int* __restrict__ row_off,
                                                     int* __restrict__ cursor,
                                                     const int* __restrict__ bsum) {
    int i = blockIdx.x * SCAN_B + threadIdx.x;
    if (i < N_NODES) {
        int r = row_off[i] + bsum[blockIdx.x];
        row_off[i] = r;
        cursor[i]  = r;
    }
}

// ---------------------------------------------------------------------------
// 4) bucket edges by destination (CSR build)
// ---------------------------------------------------------------------------
__global__ void bucket_edges(const int*   __restrict__ esrc,
                             const int*   __restrict__ edst,
                             const float* __restrict__ evals,
                             int*   __restrict__ cursor,
                             int*   __restrict__ ssrc,
                             float* __restrict__ sval) {
    int e = blockIdx.x * blockDim.x + threadIdx.x;
    if (e < N_EDGES) {
        int d   = edst[e];
        int pos = atomicAdd(&cursor[d], 1);
        ssrc[pos] = esrc[e];
        sval[pos] = evals[e];
    }
}

// ---------------------------------------------------------------------------
// 5) repack W for b64 B-fragment loads:
//    Wp2[(kq*2+lh)*256 + col] = { W[4kq+2lh][col], W[4kq+2lh+1][col] }
// ---------------------------------------------------------------------------
__global__ void pack_W(const float* __restrict__ W, float2* __restrict__ Wp2) {
    int idx = blockIdx.x * blockDim.x + threadIdx.x;   // 0..32767 exact
    int col = idx & 255;
    int r   = idx >> 8;                                // kq*2 + lh
    int k0  = r * 2;                                   // 4*kq + 2*lh
    Wp2[idx] = make_float2(W[(size_t)k0 * UNITS + col],
                           W[(size_t)(k0 + 1) * UNITS + col]);
}

// ---------------------------------------------------------------------------
// 6) gather: one wave32 per node, software-pipelined edge metadata.
//    Pure L2-bandwidth gather of x rows; zero fp atomics; one store per row.
// ---------------------------------------------------------------------------
__global__ void __launch_bounds__(256) gather_nodes(
    const float* __restrict__ x,
    const int*   __restrict__ row_off,
    const int*   __restrict__ ssrc,
    const float* __restrict__ sval,
    float*       __restrict__ conv) {
    int node = blockIdx.x * 8 + (threadIdx.x >> 5);   // grid exact: 12500*8
    int lane = threadIdx.x & 31;

    int beg = row_off[node];
    int end = row_off[node + 1];

    float4 a0 = make_float4(0.f, 0.f, 0.f, 0.f);
    float4 a1 = make_float4(0.f, 0.f, 0.f, 0.f);

    if (beg < end) {
        int   s = ssrc[beg];
        float v = sval[beg];
        for (int e = beg; e < end;) {
            const float4* xr = (const float4*)(x + (size_t)s * F_IN);
            float4 x0 = xr[lane];
            float4 x1 = xr[lane + 32];
            float  vc = v;
            ++e;
            if (e < end) { s = ssrc[e]; v = sval[e]; }   // prefetch next edge
            a0.x = fmaf(vc, x0.x, a0.x); a0.y = fmaf(vc, x0.y, a0.y);
            a0.z = fmaf(vc, x0.z, a0.z); a0.w = fmaf(vc, x0.w, a0.w);
            a1.x = fmaf(vc, x1.x, a1.x); a1.y = fmaf(vc, x1.y, a1.y);
            a1.z = fmaf(vc, x1.z, a1.z); a1.w = fmaf(vc, x1.w, a1.w);
        }
    }

    float4* cr = (float4*)(conv + (size_t)node * F_IN);
    cr[lane]      = a0;
    cr[lane + 32] = a1;
}

// ---------------------------------------------------------------------------
// 7) out = relu(conv @ W + b) via V_WMMA_F32_16X16X4_F32.
//    Wave computes 16x64; B fragments now single b64 loads from packed W.
// ---------------------------------------------------------------------------
__global__ void __launch_bounds__(256) gemm_bias_relu(
    const float* __restrict__ A,        // conv [N_NODES, F_IN]
    const v2f*   __restrict__ Wp2,      // packed W, float2[(64*2)*256]
    const float* __restrict__ bias,     // [UNITS]
    float*       __restrict__ out) {    // [N_NODES, UNITS]
    int wave  = threadIdx.x >> 5;
    int lane  = threadIdx.x & 31;
    int gwave = blockIdx.x * 8 + wave;
    int mtile = gwave >> 2;                // 0..6249
    int ng    = gwave & 3;                 // 64-col group
    int lhalf = lane >> 4;                 // 0|1
    int l16   = lane & 15;

    int row = mtile * 16 + l16;
    const float* arow = A + (size_t)row * F_IN;
    int colbase = ng * 64 + l16;

    v8f c0 = {}, c1 = {}, c2 = {}, c3 = {};

    for (int kq = 0; kq < F_IN / 4; ++kq) {
        v2f a = *(const v2f*)(arow + kq * 4 + (lhalf << 1));     // b64

        const v2f* wrow = Wp2 + ((size_t)((kq << 1) + lhalf) * UNITS + colbase);
        v2f b0 = wrow[0];                                        // b64 each
        v2f b1 = wrow[16];
        v2f b2 = wrow[32];
        v2f b3 = wrow[48];

        c0 = __builtin_amdgcn_wmma_f32_16x16x4_f32(false, a, false, b0, (short)0, c0, false, false);
        c1 = __builtin_amdgcn_wmma_f32_16x16x4_f32(false, a, false, b1, (short)0, c1, false, false);
        c2 = __builtin_amdgcn_wmma_f32_16x16x4_f32(false, a, false, b2, (short)0, c2, false, false);
        c3 = __builtin_amdgcn_wmma_f32_16x16x4_f32(false, a, false, b3, (short)0, c3, false, false);
    }

    // Fused bias + ReLU epilogue
    int orow0 = mtile * 16 + lhalf * 8;
    v8f cc[4] = { c0, c1, c2, c3 };
#pragma unroll
    for (int t = 0; t < 4; ++t) {
        int   col = ng * 64 + t * 16 + l16;
        float bv  = bias[col];
#pragma unroll
        for (int r = 0; r < 8; ++r) {
            float vv = cc[t][r] + bv;
            vv = vv > 0.f ? vv : 0.f;
            out[(size_t)(orow0 + r) * UNITS + col] = vv;
        }
    }
}

// ---------------------------------------------------------------------------
extern "C" void kernel_launch(void* const* d_in, const int* in_sizes, int n_in,
                              void* d_out, int out_size, void* d_ws, size_t ws_size,
                              hipStream_t stream) {
    const float* x     = (const float*)d_in[0];
    const int*   esrc  = (const int*)  d_in[1];
    const int*   edst  = (const int*)  d_in[2];
    const float* evals = (const float*)d_in[3];
    const float* W     = (const float*)d_in[4];
    const float* bias  = (const float*)d_in[5];
    float*       out   = (float*)d_out;

    int*    wsi     = (int*)d_ws;
    float*  conv    = (float*)d_ws + WS_CONV;
    int*    deg     = wsi + WS_DEG;
    int*    row_off = wsi + WS_ROWOFF;
    int*    cursor  = wsi + WS_CURSOR;
    int*    bsum    = wsi + WS_BSUM;
    int*    ssrc    = wsi + WS_SSRC;
    float*  sval    = (float*)(wsi + WS_SVAL);
    float2* Wp2     = (float2*)(wsi + WS_WPACK);

    zero_deg     <<<(N_NODES + 255) / 256, 256, 0, stream>>>(deg);
    count_edges  <<<N_EDGES / 256,         256, 0, stream>>>(edst, deg);
    scan_block   <<<N_SCAN_BLOCKS,      SCAN_B, 0, stream>>>(deg, row_off, bsum);
    scan_partials<<<1,                       1, 0, stream>>>(bsum, row_off);
    scan_apply   <<<N_SCAN_BLOCKS,      SCAN_B, 0, stream>>>(row_off, cursor, bsum);
    bucket_edges <<<N_EDGES / 256,         256, 0, stream>>>(esrc, edst, evals,
                                                             cursor, ssrc, sval);
    pack_W       <<<(F_IN / 4) * 2 * UNITS / 256, 256, 0, stream>>>(W, Wp2);
    gather_nodes <<<N_NODES / 8,           256, 0, stream>>>(x, row_off, ssrc, sval, conv);
    gemm_bias_relu<<<(N_NODES / 16) * 4 / 8, 256, 0, stream>>>(conv, (const v2f*)Wp2,
                                                               bias, out);
}